// RSMA_layer_21053929685157
// MI455X (gfx1250) — compile-verified
//
#include <hip/hip_runtime.h>
#include <cstdint>

#define TPB 128
#define EPSV 2e-6f

struct cplx { float x, y; };

__device__ __forceinline__ cplx cmul(cplx a, cplx b) {            // a*b
    return {a.x * b.x - a.y * b.y, a.x * b.y + a.y * b.x};
}
__device__ __forceinline__ cplx cmulc(cplx a, cplx b) {           // a*conj(b)
    return {a.x * b.x + a.y * b.y, a.y * b.x - a.x * b.y};
}
__device__ __forceinline__ void cfma_conj(cplx& acc, cplx h, cplx v) {  // acc += conj(h)*v
    acc.x += h.x * v.x + h.y * v.y;
    acc.y += h.x * v.y - h.y * v.x;
}
__device__ __forceinline__ float cabs2(cplx a) { return a.x * a.x + a.y * a.y; }
__device__ __forceinline__ cplx cscale(cplx a, float s) { return {a.x * s, a.y * s}; }

__global__ __launch_bounds__(TPB)
void rsma_kernel(const float* __restrict__ H1, const float* __restrict__ H2,
                 const float* __restrict__ H3,
                 const float* __restrict__ f1p, const float* __restrict__ f2p,
                 const float* __restrict__ f3p,
                 const float* __restrict__ r1p, const float* __restrict__ r2p,
                 const float* __restrict__ r3p,
                 const float* __restrict__ rc1p, const float* __restrict__ rc2p,
                 const float* __restrict__ rc3p,
                 const float* __restrict__ sigp, const float* __restrict__ pmaxp,
                 const float* __restrict__ V0, const float* __restrict__ V1,
                 const float* __restrict__ V2, const float* __restrict__ V3,
                 const float* __restrict__ wp, float* __restrict__ out, int Bn) {
    __shared__ float smem[3 * TPB * 16];
    const int lt = threadIdx.x;
    const int b = blockIdx.x * TPB + lt;
    if (b >= Bn) return;

    // ---- CDNA5 async global->LDS staging of H1/H2/H3 (64B per lane each) ----
    {
        uint32_t l0 = (uint32_t)(size_t)(&smem[0]);
        uint64_t g1 = (uint64_t)(size_t)H1 + (uint64_t)b * 64u;
        uint64_t g2 = (uint64_t)(size_t)H2 + (uint64_t)b * 64u;
        uint64_t g3 = (uint64_t)(size_t)H3 + (uint64_t)b * 64u;
        uint32_t l1 = l0 + (uint32_t)((0 * TPB + lt) * 64);
        uint32_t l2 = l0 + (uint32_t)((1 * TPB + lt) * 64);
        uint32_t l3 = l0 + (uint32_t)((2 * TPB + lt) * 64);
        asm volatile("global_load_async_to_lds_b128 %0, %1, off"           :: "v"(l1), "v"(g1) : "memory");
        asm volatile("global_load_async_to_lds_b128 %0, %1, off offset:16" :: "v"(l1), "v"(g1) : "memory");
        asm volatile("global_load_async_to_lds_b128 %0, %1, off offset:32" :: "v"(l1), "v"(g1) : "memory");
        asm volatile("global_load_async_to_lds_b128 %0, %1, off offset:48" :: "v"(l1), "v"(g1) : "memory");
        asm volatile("global_load_async_to_lds_b128 %0, %1, off"           :: "v"(l2), "v"(g2) : "memory");
        asm volatile("global_load_async_to_lds_b128 %0, %1, off offset:16" :: "v"(l2), "v"(g2) : "memory");
        asm volatile("global_load_async_to_lds_b128 %0, %1, off offset:32" :: "v"(l2), "v"(g2) : "memory");
        asm volatile("global_load_async_to_lds_b128 %0, %1, off offset:48" :: "v"(l2), "v"(g2) : "memory");
        asm volatile("global_load_async_to_lds_b128 %0, %1, off"           :: "v"(l3), "v"(g3) : "memory");
        asm volatile("global_load_async_to_lds_b128 %0, %1, off offset:16" :: "v"(l3), "v"(g3) : "memory");
        asm volatile("global_load_async_to_lds_b128 %0, %1, off offset:32" :: "v"(l3), "v"(g3) : "memory");
        asm volatile("global_load_async_to_lds_b128 %0, %1, off offset:48" :: "v"(l3), "v"(g3) : "memory");
    }

    // ---- Overlap: direct b128 loads of v0..v3 and per-batch scalars ----
    cplx u0[8], u1[8], u2[8], u3[8];
    {
        const float4* p0 = reinterpret_cast<const float4*>(V0 + (size_t)b * 16);
        const float4* p1 = reinterpret_cast<const float4*>(V1 + (size_t)b * 16);
        const float4* p2 = reinterpret_cast<const float4*>(V2 + (size_t)b * 16);
        const float4* p3 = reinterpret_cast<const float4*>(V3 + (size_t)b * 16);
#pragma unroll
        for (int k = 0; k < 4; ++k) {
            float4 q;
            q = p0[k]; u0[2 * k] = {q.x, q.y}; u0[2 * k + 1] = {q.z, q.w};
            q = p1[k]; u1[2 * k] = {q.x, q.y}; u1[2 * k + 1] = {q.z, q.w};
            q = p2[k]; u2[2 * k] = {q.x, q.y}; u2[2 * k + 1] = {q.z, q.w};
            q = p3[k]; u3[2 * k] = {q.x, q.y}; u3[2 * k + 1] = {q.z, q.w};
        }
    }
    const float sig = sigp[b];
    const float f1 = f1p[b], f2 = f2p[b], f3 = f3p[b];
    const float r1 = r1p[b], r2 = r2p[b], r3 = r3p[b];
    const float rce = exp2f(rc1p[b] + rc2p[b] + rc3p[b]) - 1.0f;
    const float pmax = pmaxp[b];
    float w[19];
#pragma unroll
    for (int i = 0; i < 19; ++i) w[i] = wp[i];

    // ---- wait for async LDS fill, read H back ----
#if __has_builtin(__builtin_amdgcn_s_wait_asynccnt)
    __builtin_amdgcn_s_wait_asynccnt(0);
#else
    asm volatile("s_wait_asynccnt 0" ::: "memory");
#endif
    cplx h1[8], h2[8], h3[8];
    {
        const float4* s1 = reinterpret_cast<const float4*>(&smem[(0 * TPB + lt) * 16]);
        const float4* s2 = reinterpret_cast<const float4*>(&smem[(1 * TPB + lt) * 16]);
        const float4* s3 = reinterpret_cast<const float4*>(&smem[(2 * TPB + lt) * 16]);
#pragma unroll
        for (int k = 0; k < 4; ++k) {
            float4 q;
            q = s1[k]; h1[2 * k] = {q.x, q.y}; h1[2 * k + 1] = {q.z, q.w};
            q = s2[k]; h2[2 * k] = {q.x, q.y}; h2[2 * k + 1] = {q.z, q.w};
            q = s3[k]; h3[2 * k] = {q.x, q.y}; h3[2 * k + 1] = {q.z, q.w};
        }
    }

    // ---- dots D[k][j] = Hk^H @ vj (length-8 complex) ----
    cplx D10{0,0}, D11{0,0}, D12{0,0}, D13{0,0};
    cplx D20{0,0}, D21{0,0}, D22{0,0}, D23{0,0};
    cplx D30{0,0}, D31{0,0}, D32{0,0}, D33{0,0};
#pragma unroll
    for (int t = 0; t < 8; ++t) {
        cfma_conj(D10, h1[t], u0[t]); cfma_conj(D11, h1[t], u1[t]);
        cfma_conj(D12, h1[t], u2[t]); cfma_conj(D13, h1[t], u3[t]);
        cfma_conj(D20, h2[t], u0[t]); cfma_conj(D21, h2[t], u1[t]);
        cfma_conj(D22, h2[t], u2[t]); cfma_conj(D23, h2[t], u3[t]);
        cfma_conj(D30, h3[t], u0[t]); cfma_conj(D31, h3[t], u1[t]);
        cfma_conj(D32, h3[t], u2[t]); cfma_conj(D33, h3[t], u3[t]);
    }
    const float Q10 = cabs2(D10), Q11 = cabs2(D11), Q12 = cabs2(D12), Q13 = cabs2(D13);
    const float Q20 = cabs2(D20), Q21 = cabs2(D21), Q22 = cabs2(D22), Q23 = cabs2(D23);
    const float Q30 = cabs2(D30), Q31 = cabs2(D31), Q32 = cabs2(D32), Q33 = cabs2(D33);

    // ---- z / d scalars (d's are exactly real: d = 1 + Q/S) ----
    const float iS1 = 1.0f / (sig + Q11 + Q12 + Q13);
    const float iS2 = 1.0f / (sig + Q21 + Q22 + Q23);
    const float iS3 = 1.0f / (sig + Q31 + Q32 + Q33);
    const cplx z01 = cscale(D10, iS1), z02 = cscale(D20, iS2), z03 = cscale(D30, iS3);
    const float d01 = 1.0f + Q10 * iS1, d02 = 1.0f + Q20 * iS2, d03 = 1.0f + Q30 * iS3;
    const float iT1 = 1.0f / (sig + Q12 + Q13);
    const float iT2 = 1.0f / (sig + Q21 + Q23);
    const float iT3 = 1.0f / (sig + Q31 + Q32);
    const cplx z1 = cscale(D11, iT1), z2 = cscale(D22, iT2), z3 = cscale(D33, iT3);
    const float d1 = 1.0f + Q11 * iT1, d2 = 1.0f + Q22 * iT2, d3 = 1.0f + Q33 * iT3;
    const float az1 = cabs2(z1), az2 = cabs2(z2), az3 = cabs2(z3);

    // ---- gradient coefficients (M_k @ v = H_k * (H_k^H v): rank-1) ----
    const float g01 = 2.f * w[4] / d01, g02 = 2.f * w[5] / d02, g03 = 2.f * w[6] / d03;
    const float c1 = 2.f * w[7]  * f1 / d1, a2c = 2.f * w[8]  * f2 * az2 / d2, a3c = 2.f * w[9]  * f3 * az3 / d3;
    const float c2 = 2.f * w[11] * f2 / d2, b1c = 2.f * w[12] * f1 * az1 / d1, b3c = 2.f * w[13] * f3 * az3 / d3;
    const float c3 = 2.f * w[15] * f3 / d3, e2c = 2.f * w[16] * f2 * az2 / d2, e1c = 2.f * w[17] * f1 * az1 / d1;

#pragma unroll
    for (int t = 0; t < 8; ++t) {
        // GD_v0 is real -> only real part of v0 moves
        const float gd0 = g01 * (z01.x * h1[t].x + z01.y * h1[t].y)
                        + g02 * (z02.x * h2[t].x + z02.y * h2[t].y)
                        + g03 * (z03.x * h3[t].x + z03.y * h3[t].y);
        u0[t].x += w[0] * gd0;

        const cplx t11 = cmulc(h1[t], z1);   // H1*conj(z1)
        const cplx t21 = cmul(h2[t], D21);
        const cplx t31 = cmul(h3[t], D31);
        u1[t].x += w[1] * (c1 * t11.x - a2c * t21.x - a3c * t31.x);
        u1[t].y += w[1] * (c1 * t11.y - a2c * t21.y - a3c * t31.y);

        const cplx t22 = cmulc(h2[t], z2);
        const cplx t12 = cmul(h1[t], D12);
        const cplx t32 = cmul(h3[t], D32);
        u2[t].x += w[2] * (c2 * t22.x - b1c * t12.x - b3c * t32.x);
        u2[t].y += w[2] * (c2 * t22.y - b1c * t12.y - b3c * t32.y);

        const cplx t33 = cmulc(h3[t], z3);
        const cplx t23 = cmul(h2[t], D23);
        const cplx t13 = cmul(h1[t], D13);
        u3[t].x += w[3] * (c3 * t33.x - e2c * t23.x - e1c * t13.x);
        u3[t].y += w[3] * (c3 * t33.y - e2c * t23.y - e1c * t13.y);
    }

    // ---- norms ----
    float n0sq = 0.f, n1sq = 0.f, n2sq = 0.f, n3sq = 0.f;
#pragma unroll
    for (int t = 0; t < 8; ++t) {
        n0sq += cabs2(u0[t]); n1sq += cabs2(u1[t]);
        n2sq += cabs2(u2[t]); n3sq += cabs2(u3[t]);
    }
    const float in0 = rsqrtf(n0sq), in1 = rsqrtf(n1sq), in2 = rsqrtf(n2sq), in3 = rsqrtf(n3sq);

    // ---- dots of updated vectors; q_kj = |Hk^H vju|^2 / n_j^2 ----
    cplx E10{0,0}, E11{0,0}, E12{0,0}, E13{0,0};
    cplx E20{0,0}, E21{0,0}, E22{0,0}, E23{0,0};
    cplx E30{0,0}, E31{0,0}, E32{0,0}, E33{0,0};
#pragma unroll
    for (int t = 0; t < 8; ++t) {
        cfma_conj(E10, h1[t], u0[t]); cfma_conj(E11, h1[t], u1[t]);
        cfma_conj(E12, h1[t], u2[t]); cfma_conj(E13, h1[t], u3[t]);
        cfma_conj(E20, h2[t], u0[t]); cfma_conj(E21, h2[t], u1[t]);
        cfma_conj(E22, h2[t], u2[t]); cfma_conj(E23, h2[t], u3[t]);
        cfma_conj(E30, h3[t], u0[t]); cfma_conj(E31, h3[t], u1[t]);
        cfma_conj(E32, h3[t], u2[t]); cfma_conj(E33, h3[t], u3[t]);
    }
    const float i0q = 1.0f / n0sq, i1q = 1.0f / n1sq, i2q = 1.0f / n2sq, i3q = 1.0f / n3sq;
    const float q10 = cabs2(E10) * i0q, q11 = cabs2(E11) * i1q, q12 = cabs2(E12) * i2q, q13 = cabs2(E13) * i3q;
    const float q20 = cabs2(E20) * i0q, q21 = cabs2(E21) * i1q, q22 = cabs2(E22) * i2q, q23 = cabs2(E23) * i3q;
    const float q30 = cabs2(E30) * i0q, q31 = cabs2(E31) * i1q, q32 = cabs2(E32) * i2q, q33 = cabs2(E33) * i3q;

    // ---- A4 (7x4); A = [A4 | I7] => A A^H = I7 + A4 A4^T (SPD, pinv == inv) ----
    float A[7][4];
    A[0][0] = q11;        A[0][1] = -r1 * q12;  A[0][2] = -r1 * q13;  A[0][3] = 0.f;
    A[1][0] = -r2 * q21;  A[1][1] = q22;        A[1][2] = -r2 * q23;  A[1][3] = 0.f;
    A[2][0] = -r3 * q31;  A[2][1] = -r3 * q32;  A[2][2] = q33;        A[2][3] = 0.f;
    A[3][0] = -rce * q11; A[3][1] = -rce * q12; A[3][2] = -rce * q13; A[3][3] = q10;
    A[4][0] = -rce * q21; A[4][1] = -rce * q22; A[4][2] = -rce * q23; A[4][3] = q20;
    A[5][0] = -rce * q31; A[5][1] = -rce * q32; A[5][2] = -rce * q33; A[5][3] = q30;
    A[6][0] = -10.f;      A[6][1] = -10.f;      A[6][2] = -10.f;      A[6][3] = -10.f;

    const float a4[4] = {n1sq, n2sq, n3sq, n0sq};
    const float nv[7] = {r1 * sig, r2 * sig, r3 * sig, rce * sig, rce * sig, rce * sig, -10.f * pmax};

    // b = A4 a4 - n   (7-vector RHS of the 7x7 system)
    float bv[7];
#pragma unroll
    for (int r = 0; r < 7; ++r) {
        float acc = -nv[r];
#pragma unroll
        for (int c = 0; c < 4; ++c) acc += A[r][c] * a4[c];
        bv[r] = acc;
    }

    // Woodbury: (I7 + A4 A4^T)^-1 b = b - A4 (I4 + A4^T A4)^-1 (A4^T b)
    // K = I4 + A4^T A4 (SPD, eig >= 1 -> pivot-free), augmented with c = A4^T b
    float K[4][5];
#pragma unroll
    for (int i = 0; i < 4; ++i) {
#pragma unroll
        for (int j = i; j < 4; ++j) {
            float acc = (i == j) ? 1.0f : 0.0f;
#pragma unroll
            for (int r = 0; r < 7; ++r) acc += A[r][i] * A[r][j];
            K[i][j] = acc;
        }
        float acc = 0.0f;
#pragma unroll
        for (int r = 0; r < 7; ++r) acc += A[r][i] * bv[r];
        K[i][4] = acc;
    }
#pragma unroll
    for (int i = 1; i < 4; ++i)
#pragma unroll
        for (int j = 0; j < i; ++j) K[i][j] = K[j][i];   // symmetrize

    // 4x5 Gaussian elimination + back-substitution -> y
#pragma unroll
    for (int p = 0; p < 4; ++p) {
        const float ip = 1.0f / K[p][p];
#pragma unroll
        for (int c = p; c < 5; ++c) K[p][c] *= ip;
#pragma unroll
        for (int r = p + 1; r < 4; ++r) {
            const float fr = K[r][p];
#pragma unroll
            for (int c = p; c < 5; ++c) K[r][c] -= fr * K[p][c];
        }
    }
    float y[4];
#pragma unroll
    for (int r = 3; r >= 0; --r) {
        float acc = K[r][4];
#pragma unroll
        for (int c = r + 1; c < 4; ++c) acc -= K[r][c] * y[c];
        y[r] = acc;
    }

    // lam = relu(b - A4 y)
    float lam[7];
#pragma unroll
    for (int r = 0; r < 7; ++r) {
        float acc = bv[r];
#pragma unroll
        for (int c = 0; c < 4; ++c) acc -= A[r][c] * y[c];
        lam[r] = fmaxf(acc, 0.0f);
    }

    // W[0:4] = a4 - A4^T lam
    float W4[4];
#pragma unroll
    for (int c = 0; c < 4; ++c) {
        float acc = a4[c];
#pragma unroll
        for (int r = 0; r < 7; ++r) acc -= A[r][c] * lam[r];
        W4[c] = acc;
    }
    const float s0 = in0 * sqrtf(fmaxf(W4[3], 0.f) + EPSV);
    const float s1f = in1 * sqrtf(fmaxf(W4[0], 0.f) + EPSV);
    const float s2f = in2 * sqrtf(fmaxf(W4[1], 0.f) + EPSV);
    const float s3f = in3 * sqrtf(fmaxf(W4[2], 0.f) + EPSV);

    // ---- store v0_new..v3_new (concatenated, interleaved complex) ----
    const size_t blk = (size_t)Bn * 16;
    float4* o0 = reinterpret_cast<float4*>(out + 0 * blk + (size_t)b * 16);
    float4* o1 = reinterpret_cast<float4*>(out + 1 * blk + (size_t)b * 16);
    float4* o2 = reinterpret_cast<float4*>(out + 2 * blk + (size_t)b * 16);
    float4* o3 = reinterpret_cast<float4*>(out + 3 * blk + (size_t)b * 16);
#pragma unroll
    for (int k = 0; k < 4; ++k) {
        float4 q;
        q.x = u0[2*k].x * s0;  q.y = u0[2*k].y * s0;  q.z = u0[2*k+1].x * s0;  q.w = u0[2*k+1].y * s0;  o0[k] = q;
        q.x = u1[2*k].x * s1f; q.y = u1[2*k].y * s1f; q.z = u1[2*k+1].x * s1f; q.w = u1[2*k+1].y * s1f; o1[k] = q;
        q.x = u2[2*k].x * s2f; q.y = u2[2*k].y * s2f; q.z = u2[2*k+1].x * s2f; q.w = u2[2*k+1].y * s2f; o2[k] = q;
        q.x = u3[2*k].x * s3f; q.y = u3[2*k].y * s3f; q.z = u3[2*k+1].x * s3f; q.w = u3[2*k+1].y * s3f; o3[k] = q;
    }
}

extern "C" void kernel_launch(void* const* d_in, const int* in_sizes, int n_in,
                              void* d_out, int out_size, void* d_ws, size_t ws_size,
                              hipStream_t stream) {
    (void)n_in; (void)out_size; (void)d_ws; (void)ws_size;
    const float* H1 = (const float*)d_in[0];
    const float* H2 = (const float*)d_in[1];
    const float* H3 = (const float*)d_in[2];
    const float* f1 = (const float*)d_in[3];
    const float* f2 = (const float*)d_in[4];
    const float* f3 = (const float*)d_in[5];
    const float* r1 = (const float*)d_in[6];
    const float* r2 = (const float*)d_in[7];
    const float* r3 = (const float*)d_in[8];
    const float* rc1 = (const float*)d_in[9];
    const float* rc2 = (const float*)d_in[10];
    const float* rc3 = (const float*)d_in[11];
    const float* sigma = (const float*)d_in[12];
    const float* Pmax = (const float*)d_in[13];
    const float* v0 = (const float*)d_in[14];
    const float* v1 = (const float*)d_in[15];
    const float* v2 = (const float*)d_in[16];
    const float* v3 = (const float*)d_in[17];
    // d_in[18] unique_vec (unused), d_in[19..21] H1H..H3H (recomputed via conj), d_in[22] iteration_exp (unused)
    const float* weights = (const float*)d_in[23];

    const int Bn = in_sizes[3];   // f1 is (B,1,1) float32 -> flat count == B
    const int grid = (Bn + TPB - 1) / TPB;
    rsma_kernel<<<grid, TPB, 0, stream>>>(H1, H2, H3, f1, f2, f3, r1, r2, r3,
                                          rc1, rc2, rc3, sigma, Pmax,
                                          v0, v1, v2, v3, weights,
                                          (float*)d_out, Bn);
}